// learnNavigation_89429809037650
// MI455X (gfx1250) — compile-verified
//
#include <hip/hip_runtime.h>

typedef float v2f __attribute__((ext_vector_type(2)));
typedef float v8f __attribute__((ext_vector_type(8)));

#define DEVINL __device__ __forceinline__
#define INVSQ 0.70710678118654752f

// ---------------- cross-lane primitives (VALU DPP/permlane, no LDS) --------
constexpr int dpp8imm(int a, int b, int c, int d, int e, int f, int g, int h) {
  return a | (b << 3) | (c << 6) | (d << 9) | (e << 12) | (f << 15) |
         (g << 18) | (h << 21);
}
constexpr int DPP8_XOR1 = dpp8imm(1, 0, 3, 2, 5, 4, 7, 6);
constexpr int DPP8_XOR2 = dpp8imm(2, 3, 0, 1, 6, 7, 4, 5);
constexpr int DPP8_XOR4 = dpp8imm(4, 5, 6, 7, 0, 1, 2, 3);

#if __has_builtin(__builtin_amdgcn_mov_dpp8)
DEVINL float xor1(float x) {
  return __int_as_float(
      __builtin_amdgcn_mov_dpp8(__float_as_int(x), DPP8_XOR1));
}
DEVINL float xor2(float x) {
  return __int_as_float(
      __builtin_amdgcn_mov_dpp8(__float_as_int(x), DPP8_XOR2));
}
DEVINL float xor4(float x) {
  return __int_as_float(
      __builtin_amdgcn_mov_dpp8(__float_as_int(x), DPP8_XOR4));
}
#else
DEVINL float xor1(float x) { return __shfl_xor(x, 1, 32); }
DEVINL float xor2(float x) { return __shfl_xor(x, 2, 32); }
DEVINL float xor4(float x) { return __shfl_xor(x, 4, 32); }
#endif

#if __has_builtin(__builtin_amdgcn_permlane16)
DEVINL float xor8(float x) {  // lane ^= 8 within each row of 16
  int i = __float_as_int(x);
  return __int_as_float(__builtin_amdgcn_permlane16(
      i, i, 0xFEDCBA98u, 0x76543210u, false, false));
}
#else
DEVINL float xor8(float x) { return __shfl_xor(x, 8, 32); }
#endif

#if __has_builtin(__builtin_amdgcn_permlanex16)
DEVINL float xor16(float x) {  // lane ^= 16 (identity select into other row)
  int i = __float_as_int(x);
  return __int_as_float(__builtin_amdgcn_permlanex16(
      i, i, 0x76543210u, 0xFEDCBA98u, false, false));
}
#else
DEVINL float xor16(float x) { return __shfl_xor(x, 16, 32); }
#endif

DEVINL float redhalf(float x) {  // sum across each 16-lane half
  x += xor1(x); x += xor2(x); x += xor4(x); x += xor8(x);
  return x;
}

#if __has_builtin(__builtin_amdgcn_tanhf)
DEVINL float fast_tanh(float x) { return __builtin_amdgcn_tanhf(x); }
#else
DEVINL float fast_tanh(float x) {  // v_exp based; exact saturation at +-1
  float t = __expf(2.f * x);
  return (t - 1.f) / (t + 1.f);
}
#endif

DEVINL v8f wmma4(v2f a, v2f b, v8f c) {
  return __builtin_amdgcn_wmma_f32_16x16x4_f32(false, a, false, b, (short)0, c,
                                               false, false);
}

// ---------------- WMMA wrappers ----------------
// S = (1/sqrt2) * Q2 * K2^T (rank-2, one 16x16x4 WMMA), then row-softmax.
// q/k registers hold Q2[s,j]/K2[s,j] at lane s+16j. Output P in C-layout.
// |S| <= sqrt(2) (tanh-bounded rank-2 dot), so softmax needs no max-subtract.
DEVINL void score_softmax(float q, float k, float P[8], int lane) {
  const bool lo = lane < 16;
  const float q1 = xor16(q);
  const float k1 = xor16(k);
  v2f a, b;
  a.x = lo ? q : 0.f;  a.y = lo ? q1 : 0.f;  // A: M=lane, K=0,1; K=2,3 zero
  b.x = lo ? k : 0.f;  b.y = lo ? k1 : 0.f;  // B: K=0,1 rows at N=lane
  v8f acc = {0.f, 0.f, 0.f, 0.f, 0.f, 0.f, 0.f, 0.f};
  acc = wmma4(a, b, acc);
#pragma unroll
  for (int v = 0; v < 8; ++v) {
    float e = __expf(acc[v] * INVSQ);
    P[v] = e / redhalf(e);
  }
}

// Store C-layout registers to LDS as 16x16 row-major.
DEVINL void storeC(float* buf, const float s[8], int lane) {
  const int l15 = lane & 15, hi = lane >> 4;
#pragma unroll
  for (int v = 0; v < 8; ++v) buf[(v + 8 * hi) * 16 + l15] = s[v];
}

// D = A(16x16 LDS row-major) @ B(16x16 LDS row-major); 4 chained WMMAs.
DEVINL void mm_lds_lds(const float* A, const float* B, float D[8], int lane) {
  const int l15 = lane & 15, hi = lane >> 4;
  v8f acc = {0.f, 0.f, 0.f, 0.f, 0.f, 0.f, 0.f, 0.f};
#pragma unroll
  for (int kk = 0; kk < 4; ++kk) {
    v2f a, b;
    a.x = A[l15 * 16 + 4 * kk + 2 * hi];
    a.y = A[l15 * 16 + 4 * kk + 2 * hi + 1];
    b.x = B[(4 * kk + 2 * hi) * 16 + l15];
    b.y = B[(4 * kk + 1 + 2 * hi) * 16 + l15];
    acc = wmma4(a, b, acc);
  }
#pragma unroll
  for (int v = 0; v < 8; ++v) D[v] = acc[v];
}

// D = A^T @ B  (both LDS 16x16 row-major)
DEVINL void mm_ldsT_lds(const float* A, const float* B, float D[8], int lane) {
  const int l15 = lane & 15, hi = lane >> 4;
  v8f acc = {0.f, 0.f, 0.f, 0.f, 0.f, 0.f, 0.f, 0.f};
#pragma unroll
  for (int kk = 0; kk < 4; ++kk) {
    v2f a, b;
    a.x = A[(4 * kk + 2 * hi) * 16 + l15];
    a.y = A[(4 * kk + 1 + 2 * hi) * 16 + l15];
    b.x = B[(4 * kk + 2 * hi) * 16 + l15];
    b.y = B[(4 * kk + 1 + 2 * hi) * 16 + l15];
    acc = wmma4(a, b, acc);
  }
#pragma unroll
  for (int v = 0; v < 8; ++v) D[v] = acc[v];
}

// Build one B-operand VGPR (rows rA for lanes 0-15, rB for lanes 16-31,
// value at column = lane&15) from a matrix held in C-layout registers.
template <int rA, int rB>
DEVINL float bfromC(const float (&s)[8], int lane) {
  float ra, rb;
  if constexpr (rA < 8) ra = s[rA]; else ra = xor16(s[rA - 8]);
  if constexpr (rB >= 8) rb = s[rB - 8]; else rb = xor16(s[rB]);
  return (lane < 16) ? ra : rb;
}

// D = W(global 16x16 row-major) @ S(C-layout registers); 4 chained WMMAs.
DEVINL void mm_w_creg(const float* __restrict__ W, const float (&S)[8],
                      float D[8], int lane) {
  const int l15 = lane & 15, hi = lane >> 4;
  v8f acc = {0.f, 0.f, 0.f, 0.f, 0.f, 0.f, 0.f, 0.f};
#define PHN_CHUNK(kk)                                                  \
  {                                                                    \
    v2f a, b;                                                          \
    a.x = W[l15 * 16 + 4 * kk + 2 * hi];                               \
    a.y = W[l15 * 16 + 4 * kk + 2 * hi + 1];                           \
    b.x = bfromC<4 * kk, 4 * kk + 2>(S, lane);                         \
    b.y = bfromC<4 * kk + 1, 4 * kk + 3>(S, lane);                     \
    acc = wmma4(a, b, acc);                                            \
  }
  PHN_CHUNK(0) PHN_CHUNK(1) PHN_CHUNK(2) PHN_CHUNK(3)
#undef PHN_CHUNK
#pragma unroll
  for (int v = 0; v < 8; ++v) D[v] = acc[v];
}

// ---------------- R / J branch ----------------
DEVINL void branchRJ(const float* __restrict__ Aq, const float* __restrict__ Ak,
                     const float* __restrict__ Av, const float* __restrict__ W,
                     float x0, float x1, float x2, float x3, float* bufA,
                     float* bufP, float* dest, bool square, int i, int lane) {
  const int l15 = lane & 15, hi = lane >> 4;
  const float q = fast_tanh(Aq[l15 * 4] * x0 + Aq[l15 * 4 + 1] * x1 +
                            Aq[l15 * 4 + 2] * x2 + Aq[l15 * 4 + 3] * x3);
  const float k = fast_tanh(Ak[l15 * 4] * x0 + Ak[l15 * 4 + 1] * x1 +
                            Ak[l15 * 4 + 2] * x2 + Ak[l15 * 4 + 3] * x3);
  const float vv = fast_tanh(Av[l15 * 4] * x0 + Av[l15 * 4 + 1] * x1 +
                             Av[l15 * 4 + 2] * x2 + Av[l15 * 4 + 3] * x3);
  float P[8];
  score_softmax(q, k, P, lane);
  storeC(bufA, P, lane);             // P -> A-operand staging
  bufP[l15 * 16 + hi] = vv;          // V2 -> zero-padded B tile (cols 0,1)
  float o2[8];
  mm_lds_lds(bufA, bufP, o2, lane);  // o2 = P @ V2pad (cols>=2 stay exact 0)
  float M[8];
  mm_w_creg(W, o2, M, lane);         // M = Ao @ o2
  if (l15 < 2) {                     // scatter 16x2 non-zeros into 96x96
    const int n = l15 ? (i + 1) % 24 : (i + 23) % 24;
#pragma unroll
    for (int v = 0; v < 8; ++v) {
      const int h = v + 8 * hi;
      float t = fast_tanh(M[v]);
      float val = square ? t * t : t;
      const int flat = h * 24 + n;
      const int r = ((flat >> 3) & 1) * 48 + 2 * i + ((flat >> 1) & 1);
      const int c = ((flat >> 2) & 1) * 48 + 2 * (flat >> 4) + (flat & 1);
      dest[r * 96 + c] = val;
    }
  }
}

// ---------------- H branch (analytic gradient) ----------------
DEVINL void branchH(const float* __restrict__ Aq, const float* __restrict__ Ak,
                    const float* __restrict__ Av, const float* __restrict__ Ao,
                    float x0, float x1, float x2, float x3, float* bufA,
                    float* bufP, float* gL, int i, int lane) {
  const int l15 = lane & 15, hi = lane >> 4;
  float aqr[4], akr[4], avr[4];
#pragma unroll
  for (int c = 0; c < 4; ++c) {
    aqr[c] = Aq[l15 * 4 + c];
    akr[c] = Ak[l15 * 4 + c];
    avr[c] = Av[l15 * 4 + c];
  }
  const float q = fast_tanh(aqr[0] * x0 + aqr[1] * x1 + aqr[2] * x2 + aqr[3] * x3);
  const float k = fast_tanh(akr[0] * x0 + akr[1] * x1 + akr[2] * x2 + akr[3] * x3);
  const float vv = fast_tanh(avr[0] * x0 + avr[1] * x1 + avr[2] * x2 + avr[3] * x3);
  float P[8];
  score_softmax(q, k, P, lane);
  float aov[8];
#pragma unroll
  for (int v = 0; v < 8; ++v) aov[v] = (lane < 16) ? Ao[v] : Ao[v + 8];
  // w[s] = sum_r P[r,s]*Ao[r]   (half-mirrored)
  float wpart = 0.f;
#pragma unroll
  for (int v = 0; v < 8; ++v) wpart += P[v] * aov[v];
  const float wm = wpart + xor16(wpart);
  // z[j] = sum_s w[s]*V2[s,j] ; s = silu(z0)+silu(z1)
  const float zj = redhalf(vv * wm);
  const float sig = 1.f / (1.f + __expf(-zj));
  const float sil = zj * sig;
  const float ssc = sil + xor16(sil);
  const float dzj = 2.f * ssc * (sig + zj * sig * (1.f - sig));
  const float sumdz = dzj + xor16(dzj) + 22.f * ssc;  // dense dz; off-support = s
  // u[s] = V2[s,0]dz0 + V2[s,1]dz1
  float um = vv * dzj;
  um = um + xor16(um);
  // dS[r,s] = inv * Ao[r] * P[r,s] * (u[s] - mu[r])
  float dS[8];
#pragma unroll
  for (int v = 0; v < 8; ++v) {
    float mu = redhalf(P[v] * um);
    dS[v] = INVSQ * aov[v] * P[v] * (um - mu);
  }
  storeC(bufA, dS, lane);
  bufP[l15 * 16 + hi] = k;
  float dq[8];
  mm_lds_lds(bufA, bufP, dq, lane);   // dQ2 = dS @ K2pad
  bufP[l15 * 16 + hi] = q;
  float dk[8];
  mm_ldsT_lds(bufA, bufP, dk, lane);  // dK2 = dS^T @ Q2pad
  // relayout dQ/dK (C-layout, cols 0,1 valid) to vector layout via bufA
  storeC(bufA, dq, lane);
  const float dqv = bufA[l15 * 16 + hi];
  storeC(bufA, dk, lane);
  const float dkv = bufA[l15 * 16 + hi];
  float tq = (1.f - q * q) * dqv;  tq += xor16(tq);  // sum over j, mirrored
  float tk = (1.f - k * k) * dkv;  tk += xor16(tk);
  // V-path: tv[s] = w[s]*(sum_dz - dz0*V[s,0]^2 - dz1*V[s,1]^2)
  float sq = vv * vv * dzj;
  sq = sq + xor16(sq);
  const float tv = wm * (sumdz - sq);
  // g_c = sum_s Aq[s,c]*tq[s] + Ak[s,c]*tk[s] + Av[s,c]*tv[s]  (one redhalf)
  float g4[4];
#pragma unroll
  for (int c = 0; c < 4; ++c)
    g4[c] = redhalf(aqr[c] * tq + akr[c] * tk + avr[c] * tv);
  if (lane < 4)
    gL[4 * i + lane] =
        (lane == 0) ? g4[0] : (lane == 1) ? g4[1] : (lane == 2) ? g4[2] : g4[3];
}

__global__ __launch_bounds__(768, 1) void phnet_kernel(
    const float* __restrict__ inputs, const float* __restrict__ HAq,
    const float* __restrict__ HAk, const float* __restrict__ HAv,
    const float* __restrict__ HAo, const float* __restrict__ JAq,
    const float* __restrict__ JAk, const float* __restrict__ JAv,
    const float* __restrict__ JAo, const float* __restrict__ RAq,
    const float* __restrict__ RAk, const float* __restrict__ RAv,
    const float* __restrict__ RAo, float* __restrict__ out) {
  extern __shared__ float smem[];
  float* Ju = smem;                // 9216
  float* Ru = Ju + 9216;           // 9216
  float* inL = Ru + 9216;          // 96
  float* gL = inL + 96;            // 96
  float* bufA0 = gL + 96;          // 24*256
  float* bufP0 = bufA0 + 24 * 256; // 24*256 (zero-padded B tiles)

  const int tid = threadIdx.x;
  const int b = blockIdx.x;
  const int lane = tid & 31;
  const int wv = tid >> 5;  // sample/agent index i, one wave each
  const int hi = lane >> 4;

  __builtin_prefetch(inputs + (size_t)b * 192, 0, 1);

  for (int idx = tid; idx < 2 * 9216; idx += 768) Ju[idx] = 0.f;
  for (int idx = tid; idx < 24 * 256; idx += 768) bufP0[idx] = 0.f;
  if (tid < 96)
    inL[tid] =
        inputs[(size_t)b * 192 + tid] - inputs[(size_t)b * 192 + 96 + tid];
  __syncthreads();

  float* bufA = bufA0 + wv * 256;
  float* bufP = bufP0 + wv * 256;
  const int i = wv;
  const int nj = hi ? (i + 1) % 24 : (i + 23) % 24;
  // x2[:,j] = posvel[i] - posvel[nj]   (LL[i, i+-1] = -1)
  const float x0 = inL[2 * i] - inL[2 * nj];
  const float x1 = inL[2 * i + 1] - inL[2 * nj + 1];
  const float x2 = inL[48 + 2 * i] - inL[48 + 2 * nj];
  const float x3 = inL[48 + 2 * i + 1] - inL[48 + 2 * nj + 1];

  branchRJ(RAq, RAk, RAv, RAo, x0, x1, x2, x3, bufA, bufP, Ru, true, i, lane);
  branchRJ(JAq, JAk, JAv, JAo, x0, x1, x2, x3, bufA, bufP, Ju, false, i, lane);
  branchH(HAq, HAk, HAv, HAo, x0, x1, x2, x3, bufA, bufP, gL, i, lane);

  __syncthreads();

  // dx = (Jmat - Rmat) @ g, 96 rows handled by threads 0..95
  if (tid < 96) {
    const int r = tid;
    const float gr = gL[r];
    float acc = 0.f, rowsum = 0.f;
    for (int c = 0; c < 96; ++c) {
      float ju_rc = Ju[r * 96 + c], ju_cr = Ju[c * 96 + r];
      float rs = Ru[r * 96 + c] + Ru[c * 96 + r];
      if (c != r) {
        acc += (ju_rc - ju_cr + rs) * gL[c];
        rowsum += rs;
      }
    }
    acc -= (2.f * Ru[r * 96 + r] + rowsum) * gr;
    if (r < 48) {
      float dd = inputs[(size_t)b * 192 + 144 + r];
      out[(size_t)b * 192 + r] = acc + dd;
      out[(size_t)b * 192 + 96 + r] = dd;
      out[(size_t)b * 192 + 144 + r] = 0.f;
    } else {
      out[(size_t)b * 192 + r] = acc;
    }
  }
}

extern "C" void kernel_launch(void* const* d_in, const int* in_sizes, int n_in,
                              void* d_out, int out_size, void* d_ws,
                              size_t ws_size, hipStream_t stream) {
  (void)n_in; (void)d_ws; (void)ws_size; (void)out_size;
  const float* inputs = (const float*)d_in[0];
  const float* HAq = (const float*)d_in[1];
  const float* HAk = (const float*)d_in[2];
  const float* HAv = (const float*)d_in[3];
  const float* HAo = (const float*)d_in[4];
  const float* JAq = (const float*)d_in[5];
  const float* JAk = (const float*)d_in[6];
  const float* JAv = (const float*)d_in[7];
  const float* JAo = (const float*)d_in[8];
  const float* RAq = (const float*)d_in[9];
  const float* RAk = (const float*)d_in[10];
  const float* RAv = (const float*)d_in[11];
  const float* RAo = (const float*)d_in[12];
  float* out = (float*)d_out;

  const int batch = in_sizes[0] / 192;
  const size_t shmem =
      (size_t)(2 * 9216 + 96 + 96 + 2 * 24 * 256) * sizeof(float);
  hipFuncSetAttribute((const void*)phnet_kernel,
                      hipFuncAttributeMaxDynamicSharedMemorySize, (int)shmem);
  phnet_kernel<<<batch, 768, shmem, stream>>>(inputs, HAq, HAk, HAv, HAo, JAq,
                                              JAk, JAv, JAo, RAq, RAk, RAv,
                                              RAo, out);
}